// DRCapsuleFC_79817672229002
// MI455X (gfx1250) — compile-verified
//
#include <hip/hip_runtime.h>

typedef __attribute__((ext_vector_type(2))) float v2f;
typedef __attribute__((ext_vector_type(8))) float v8f;

#define B_    128
#define N_    2048
#define A_    16
#define M_    64
#define D_    16
#define KTOT  (N_*A_)   /* 32768 */
#define MD    (M_*D_)   /* 1024  */
#define KC    512       /* split-K chunk */
#define KCH   (KTOT/KC) /* 64 chunks */
#define LDX   (KC+2)    /* LDS pad: stride%64==2 -> conflict-free b64 reads */
#define LN_EPS 1e-5f

// ---------------------------------------------------------------- kernel 0
__global__ void zero_acc_kernel(float* __restrict__ acc) {
    int i = blockIdx.x * blockDim.x + threadIdx.x;
    if (i < B_ * MD) acc[i] = 0.0f;
}

// ---------------------------------------------------------------- kernel 1
// acc[b, m*16+d] += (1/64) * sum_k X[b,k] * W[k, m*16+d]   (split-K, atomic)
// grid = (8 b-tiles, 64 k-chunks), block = 256 (8 waves); wave w owns m=8w..8w+7
__global__ __launch_bounds__(256) void gemm1_kernel(const float* __restrict__ X,
                                                    const float* __restrict__ Wg,
                                                    float* __restrict__ acc) {
    __shared__ __align__(16) float xs[16 * LDX];
    const int tid   = threadIdx.x;
    const int b0    = blockIdx.x * 16;
    const int kbase = blockIdx.y * KC;

    // coalesced cooperative load of the 16 x KC X-tile
    for (int i = tid; i < 16 * KC; i += 256) {
        int r = i >> 9, c = i & (KC - 1);
        xs[r * LDX + c] = X[(size_t)(b0 + r) * KTOT + kbase + c];
    }
    __syncthreads();

    const int w = tid >> 5, l = tid & 31;
    const int h = l >> 4, col = l & 15;          // col = N index; also A-row = l&15
    const float* xp = xs + (size_t)(l & 15) * LDX + 2 * h;

    for (int mi = 0; mi < 8; ++mi) {
        const int m = w * 8 + mi;
        v8f c8 = {0.f,0.f,0.f,0.f,0.f,0.f,0.f,0.f};
        const float* wp = Wg + (size_t)(kbase + 2 * h) * MD + m * 16 + col;
        #pragma unroll 4
        for (int k0 = 0; k0 < KC; k0 += 4) {
            v2f a = *(const v2f*)(xp + k0);      // ds_load_b64, bank-conflict-free
            v2f b;
            b.x = wp[(size_t)k0 * MD];           // B[k0+2h,   col], 64B coalesced/half
            b.y = wp[(size_t)k0 * MD + MD];      // B[k0+2h+1, col]
            c8 = __builtin_amdgcn_wmma_f32_16x16x4_f32(
                     false, a, false, b, (short)0, c8, false, false);
        }
        const float scale = 1.0f / 64.0f;        // uniform routing coeff 1/M
        #pragma unroll
        for (int j = 0; j < 8; ++j)              // C/D layout: row = j + 8h, col = l&15
            atomicAdd(&acc[(size_t)(b0 + j + 8 * h) * MD + m * 16 + col], c8[j] * scale);
    }
}

// ---------------------------------------------------------------- kernel 2
// squash over d, emit squashed v (ws) + LayerNorm'd v (d_out tail)
__global__ void squash_ln_kernel(const float* __restrict__ acc,
                                 const float* __restrict__ gamma,
                                 const float* __restrict__ beta,
                                 float* __restrict__ vsq,
                                 float* __restrict__ out_v) {
    int idx = blockIdx.x * blockDim.x + threadIdx.x;  // (b*64 + m)
    if (idx >= B_ * M_) return;
    float x[16]; float sq = 0.f;
    #pragma unroll
    for (int d = 0; d < 16; ++d) { x[d] = acc[idx * 16 + d]; sq += x[d] * x[d]; }
    float coef = (sq / (1.f + sq)) * rsqrtf(sq);
    float mu = 0.f;
    #pragma unroll
    for (int d = 0; d < 16; ++d) { x[d] *= coef; vsq[idx * 16 + d] = x[d]; mu += x[d]; }
    mu *= (1.f / 16.f);
    float var = 0.f;
    #pragma unroll
    for (int d = 0; d < 16; ++d) { float t = x[d] - mu; var += t * t; }
    var *= (1.f / 16.f);
    float inv = rsqrtf(var + LN_EPS);
    #pragma unroll
    for (int d = 0; d < 16; ++d)
        out_v[idx * 16 + d] = (x[d] - mu) * inv * gamma[d] + beta[d];
}

// ---------------------------------------------------------------- kernel 3
// new_qk[b,n,m] = sum_{a,d} X[b,n,a] * w[n,a,m,d] * vsq[b,m,d]
// grid = N_ blocks (one n each; w[n] re-read hits L2 since w fits in 192MB L2)
// block = 256; wave w owns m=8w..8w+7, loops b-tiles with B-frags register-resident
__global__ __launch_bounds__(256) void gemm2_kernel(const float* __restrict__ X,
                                                    const float* __restrict__ Wg,
                                                    const float* __restrict__ vsq,
                                                    float* __restrict__ qk) {
    __shared__ __align__(16) float xs[B_ * 18];  // 128 x 16, pad-18 (even, 18%64=18)
    const int n   = blockIdx.x;
    const int tid = threadIdx.x;
    for (int i = tid; i < B_ * A_; i += 256) {
        int r = i >> 4, a = i & 15;
        xs[r * 18 + a] = X[(size_t)r * KTOT + n * 16 + a];
    }
    __syncthreads();

    const int w = tid >> 5, l = tid & 31;
    const int h = l >> 4, col = l & 15;
    const size_t woff = (size_t)n * (A_ * MD);

    for (int mi = 0; mi < 8; ++mi) {
        const int m = w * 8 + mi;
        v2f bf[4];
        #pragma unroll
        for (int kc = 0; kc < 4; ++kc) {         // B[a, d] = w[n, a, m, d], K=a
            size_t o = woff + (size_t)(kc * 4 + 2 * h) * MD + m * 16 + col;
            bf[kc].x = Wg[o];
            bf[kc].y = Wg[o + MD];
        }
        for (int bt = 0; bt < 8; ++bt) {
            v8f c8 = {0.f,0.f,0.f,0.f,0.f,0.f,0.f,0.f};
            const float* xp = xs + (size_t)(bt * 16 + (l & 15)) * 18 + 2 * h;
            #pragma unroll
            for (int kc = 0; kc < 4; ++kc) {
                v2f a = *(const v2f*)(xp + kc * 4);
                c8 = __builtin_amdgcn_wmma_f32_16x16x4_f32(
                         false, a, false, bf[kc], (short)0, c8, false, false);
            }
            // weight by vsq[b,m,d] (d = column = lane&15) and reduce across d
            #pragma unroll
            for (int j = 0; j < 8; ++j) {
                const int b = bt * 16 + j + 8 * h;
                float t = c8[j] * vsq[((size_t)b * M_ + m) * 16 + col];
                t += __shfl_xor(t, 1);           // xor<=8 stays inside 16-lane half
                t += __shfl_xor(t, 2);
                t += __shfl_xor(t, 4);
                t += __shfl_xor(t, 8);
                if (col == 0)
                    qk[(size_t)b * (N_ * M_) + (size_t)n * M_ + m] = t;
            }
        }
    }
}

// ----------------------------------------------------------------
extern "C" void kernel_launch(void* const* d_in, const int* in_sizes, int n_in,
                              void* d_out, int out_size, void* d_ws, size_t ws_size,
                              hipStream_t stream) {
    const float* input = (const float*)d_in[0];   // (128, 2048, 16)
    const float* wg    = (const float*)d_in[1];   // (2048, 16, 64, 16)
    const float* gamma = (const float*)d_in[2];   // (16,)
    const float* beta  = (const float*)d_in[3];   // (16,)

    float* out_qk = (float*)d_out;                        // (128, 2048, 64)
    float* out_v  = out_qk + (size_t)B_ * N_ * M_;        // (128, 64, 16)

    float* acc = (float*)d_ws;                            // 128*1024 floats (512 KB)
    float* vsq = acc + (size_t)B_ * MD;                   // 128*1024 floats (512 KB)

    zero_acc_kernel<<<dim3((B_ * MD + 255) / 256), dim3(256), 0, stream>>>(acc);
    gemm1_kernel<<<dim3(8, KCH), dim3(256), 0, stream>>>(input, wg, acc);
    squash_ln_kernel<<<dim3((B_ * M_ + 255) / 256), dim3(256), 0, stream>>>(
        acc, gamma, beta, vsq, out_v);
    gemm2_kernel<<<dim3(N_), dim3(256), 0, stream>>>(input, wg, vsq, out_qk);
}